// GCNChain_23167053595560
// MI455X (gfx1250) — compile-verified
//
#include <hip/hip_runtime.h>
#include <math.h>

typedef __attribute__((ext_vector_type(2))) float v2f;
typedef __attribute__((ext_vector_type(8))) float v8f;

#define D_H 128

// ---------------------------------------------------------------- degree ----
__global__ void deg_init_kernel(float* __restrict__ deg, int n) {
  int i = blockIdx.x * blockDim.x + threadIdx.x;
  if (i < n) deg[i] = 1.0f;  // self loop contributes 1 to every node
}

__global__ void deg_count_kernel(const int* __restrict__ tgt,
                                 float* __restrict__ deg, int e) {
  int i = blockIdx.x * blockDim.x + threadIdx.x;
  if (i < e)
    __hip_atomic_fetch_add(&deg[tgt[i]], 1.0f, __ATOMIC_RELAXED,
                           __HIP_MEMORY_SCOPE_AGENT);
}

__global__ void dinv_kernel(float* __restrict__ deg, int n) {
  int i = blockIdx.x * blockDim.x + threadIdx.x;
  if (i < n) {
    float d = deg[i];
    deg[i] = (d > 0.0f) ? rsqrtf(d) : 0.0f;  // overwrite with D^{-1/2}
  }
}

// ------------------------------------------------------- XW = X @ W (WMMA) --
// One wave32 computes one 16x16 tile of XW with V_WMMA_F32_16X16X4_F32,
// stepping K by 4 (32 WMMAs, fully unrolled). fp32 in, fp32 accumulate.
__global__ void __launch_bounds__(256)
gemm_wmma_kernel(const float* __restrict__ X, const float* __restrict__ W,
                 float* __restrict__ XW, int n) {
  const int wave  = threadIdx.x >> 5;
  const int lane  = threadIdx.x & 31;
  const int tile  = blockIdx.x * 8 + wave;   // 8 waves per block
  const int tileM = tile >> 3;               // n/16 row tiles
  const int tileN = tile & 7;                // 128/16 = 8 col tiles
  const int half  = lane >> 4;               // 0: lanes 0-15, 1: lanes 16-31
  const int l15   = lane & 15;
  const int row   = tileM * 16 + l15;        // A-matrix M = l15
  const int col   = tileN * 16 + l15;        // B/C/D N = l15

  // A lane base: X[row][half*2], pairs {K, K+1}
  const float* ap = X + (size_t)row * D_H + half * 2;
  // B lane base: W[half*2][col], pairs {K, K+1} down column `col`
  const float* bp = W + (size_t)(half * 2) * D_H + col;

  v8f acc = {};
#pragma unroll
  for (int k = 0; k < D_H; k += 4) {
    v2f a, b;
    a.x = ap[k];            // A[row][k + 2*half]
    a.y = ap[k + 1];        // A[row][k + 2*half + 1]
    b.x = bp[k * D_H];      // W[k + 2*half][col]
    b.y = bp[k * D_H + D_H];// W[k + 2*half + 1][col]
    acc = __builtin_amdgcn_wmma_f32_16x16x4_f32(
        /*neg_a=*/false, a, /*neg_b=*/false, b,
        /*c_mod=*/(short)0, acc, /*reuse_a=*/false, /*reuse_b=*/false);
  }

  // C/D layout: VGPR r -> M = r (lanes 0-15) / M = r + 8 (lanes 16-31)
  float* op = XW + (size_t)(tileM * 16 + half * 8) * D_H + col;
#pragma unroll
  for (int r = 0; r < 8; ++r) op[(size_t)r * D_H] = acc[r];
}

// ------------------------------------- out = b + dinv[i]^2 * XW[i]  (self) --
__global__ void out_init_kernel(const float* __restrict__ XW,
                                const float* __restrict__ dinv,
                                const float* __restrict__ b,
                                float* __restrict__ out, int n) {
  int idx = blockIdx.x * blockDim.x + threadIdx.x;
  if (idx < n * D_H) {
    int node = idx >> 7;
    float di = dinv[node];
    out[idx] = b[idx & (D_H - 1)] + di * di * XW[idx];
  }
}

// ----------------------------- scatter: out[tgt] += norm * XW[src] ----------
// One wave per edge; each lane handles 4 dims (float4 gather, 4 f32 atomics).
__global__ void __launch_bounds__(256)
scatter_kernel(const float* __restrict__ XW, const float* __restrict__ dinv,
               const int* __restrict__ src, const int* __restrict__ tgt,
               float* __restrict__ out, int e) {
  int gw = (blockIdx.x * blockDim.x + threadIdx.x) >> 5;
  if (gw >= e) return;
  int lane = threadIdx.x & 31;
  int s = src[gw];
  int t = tgt[gw];
  float norm = dinv[s] * dinv[t];
  float4 v = ((const float4*)(XW + (size_t)s * D_H))[lane];
  float* o = out + (size_t)t * D_H + lane * 4;
  __hip_atomic_fetch_add(o + 0, norm * v.x, __ATOMIC_RELAXED,
                         __HIP_MEMORY_SCOPE_AGENT);
  __hip_atomic_fetch_add(o + 1, norm * v.y, __ATOMIC_RELAXED,
                         __HIP_MEMORY_SCOPE_AGENT);
  __hip_atomic_fetch_add(o + 2, norm * v.z, __ATOMIC_RELAXED,
                         __HIP_MEMORY_SCOPE_AGENT);
  __hip_atomic_fetch_add(o + 3, norm * v.w, __ATOMIC_RELAXED,
                         __HIP_MEMORY_SCOPE_AGENT);
}

// ------------------- MessageNorm + GELU: one wave per node ------------------
__device__ __forceinline__ float gelu_exact(float v) {
  return 0.5f * v * (1.0f + erff(v * 0.70710678118654752f));
}

__global__ void __launch_bounds__(256)
finalize_kernel(const float* __restrict__ X, const float* __restrict__ scale,
                float* __restrict__ out, int n) {
  int gw = (blockIdx.x * blockDim.x + threadIdx.x) >> 5;
  if (gw >= n) return;
  int lane = threadIdx.x & 31;
  float4 m = ((const float4*)(out + (size_t)gw * D_H))[lane];
  float4 x = ((const float4*)(X + (size_t)gw * D_H))[lane];
  float sm = m.x * m.x + m.y * m.y + m.z * m.z + m.w * m.w;
  float sx = x.x * x.x + x.y * x.y + x.z * x.z + x.w * x.w;
#pragma unroll
  for (int off = 16; off > 0; off >>= 1) {   // wave32 butterfly reduction
    sm += __shfl_xor(sm, off, 32);
    sx += __shfl_xor(sx, off, 32);
  }
  float denom = fmaxf(sqrtf(sm), 1e-12f);
  float g = sqrtf(sx) * scale[0] / denom;
  float4 r;
  r.x = gelu_exact(m.x * g);
  r.y = gelu_exact(m.y * g);
  r.z = gelu_exact(m.z * g);
  r.w = gelu_exact(m.w * g);
  ((float4*)(out + (size_t)gw * D_H))[lane] = r;
}

// ---------------------------------------------------------------------------
extern "C" void kernel_launch(void* const* d_in, const int* in_sizes, int n_in,
                              void* d_out, int out_size, void* d_ws,
                              size_t ws_size, hipStream_t stream) {
  const float* X     = (const float*)d_in[0];
  const int*   ei    = (const int*)d_in[1];
  const float* W     = (const float*)d_in[2];
  const float* b     = (const float*)d_in[3];
  const float* scale = (const float*)d_in[4];

  const int n = in_sizes[0] / D_H;   // 100000
  const int e = in_sizes[1] / 2;     // 3200000
  const int* src = ei;               // edge_index[0]
  const int* tgt = ei + e;           // edge_index[1]

  float* ws   = (float*)d_ws;
  float* XW   = ws;                          // n * 128 floats
  float* dinv = ws + (size_t)n * D_H;        // n floats (deg then D^{-1/2})
  float* out  = (float*)d_out;

  // 1) degree (with self loops), then D^{-1/2} in place
  deg_init_kernel<<<(n + 255) / 256, 256, 0, stream>>>(dinv, n);
  deg_count_kernel<<<(e + 255) / 256, 256, 0, stream>>>(tgt, dinv, e);
  dinv_kernel<<<(n + 255) / 256, 256, 0, stream>>>(dinv, n);

  // 2) XW = X @ W via fp32 WMMA (n divisible by 16: 6250 row tiles)
  gemm_wmma_kernel<<<n / 16, 256, 0, stream>>>(X, W, XW, n);

  // 3) out = b + self-loop term
  out_init_kernel<<<(n * D_H + 255) / 256, 256, 0, stream>>>(XW, dinv, b, out, n);

  // 4) edge scatter-add (one wave per edge)
  scatter_kernel<<<(e * 32 + 255) / 256, 256, 0, stream>>>(XW, dinv, src, tgt,
                                                           out, e);

  // 5) MessageNorm + exact GELU (one wave per node)
  finalize_kernel<<<(n + 7) / 8, 256, 0, stream>>>(X, scale, out, n);
}